// ManifoldRenderer_44263932952601
// MI455X (gfx1250) — compile-verified
//
#include <hip/hip_runtime.h>
#include <math.h>

typedef float v2f __attribute__((ext_vector_type(2)));
typedef float v8f __attribute__((ext_vector_type(8)));

#define R_SPHERE 3.0f
#define NEARP    0.0f
#define FARP     100.0f
#define EPSI     1e-4f
#define R0C      1.0f
#define NS       64
#define NL       4
#define NH       64

// CDNA5 hardware transcendental tanh (TRANS32 pipe, co-executes with WMMA).
__device__ __forceinline__ float fast_tanh(float x) {
#if __has_builtin(__builtin_amdgcn_tanhf)
  return __builtin_amdgcn_tanhf(x);
#else
  float y;
  asm("v_tanh_f32 %0, %1" : "=v"(y) : "v"(x));
  return y;
#endif
}

// One 16-point MLP tile: D = [p,1](16x4) @ [W1;b1](4x64), tanh, dot W2.
// Returns per-group (16-lane) reduced sums: lanes 0-15 share sum[v] for point m=v,
// lanes 16-31 share sum[v] for point m=v+8.
__device__ __forceinline__ void mlp_wmma(v2f a, const v2f Bw[4], const float w2r[4],
                                         float sum[8]) {
  float part[8];
#pragma unroll
  for (int v = 0; v < 8; ++v) part[v] = 0.0f;
#pragma unroll
  for (int j = 0; j < 4; ++j) {
    v8f c = {0.0f, 0.0f, 0.0f, 0.0f, 0.0f, 0.0f, 0.0f, 0.0f};
    // D = A(16x4 f32) x B(4x16 f32) + C ; full fp32, K=4 == [x,y,z,bias]
    c = __builtin_amdgcn_wmma_f32_16x16x4_f32(false, a, false, Bw[j],
                                              (short)0, c, false, false);
#pragma unroll
    for (int v = 0; v < 8; ++v) part[v] += fast_tanh(c[v]) * w2r[j];
  }
#pragma unroll
  for (int off = 1; off <= 8; off <<= 1) {
#pragma unroll
    for (int v = 0; v < 8; ++v) part[v] += __shfl_xor(part[v], off, 32);
  }
#pragma unroll
  for (int v = 0; v < 8; ++v) sum[v] = part[v];
}

__global__ __launch_bounds__(256) void manifold_march_kernel(
    const float* __restrict__ rays_o, const float* __restrict__ rays_d,
    const float* __restrict__ levels, const float* __restrict__ W1,
    const float* __restrict__ b1, const float* __restrict__ W2,
    const float* __restrict__ b2, float* __restrict__ out_dist,
    float* __restrict__ out_mask, int Nrays) {
  // Per-wave-private LDS regions (no cross-wave sharing -> only dscnt waits).
  __shared__ float         scalLDS[8][16][NS];  // 32 KB
  __shared__ unsigned char valLDS[8][16][NS];   //  8 KB
  __shared__ float         tmpLDS[8][NL][16];   //  2 KB
  __shared__ float         diLDS[8][16][NL];    //  2 KB
  __shared__ float         miLDS[8][16][NL];    //  2 KB

  const int tid  = threadIdx.x;
  const int w    = tid >> 5;
  const int lane = tid & 31;
  const int hf   = lane >> 4;  // which half of the wave
  const int m    = lane & 15;  // local ray / WMMA row

  int g = blockIdx.x * 128 + w * 16 + m;
  if (g >= Nrays) g = Nrays - 1;  // uniform clamp keeps EXEC all-1s

  const float ox = rays_o[3 * g + 0], oy = rays_o[3 * g + 1], oz = rays_o[3 * g + 2];
  const float dx = rays_d[3 * g + 0], dy = rays_d[3 * g + 1], dz = rays_d[3 * g + 2];

  // Sphere bounds
  const float bq   = ox * dx + oy * dy + oz * dz;
  const float cq   = ox * ox + oy * oy + oz * oz - R_SPHERE * R_SPHERE;
  const float disc = bq * bq - cq;
  const bool  hit  = disc > 0.0f;
  const float sq   = sqrtf(hit ? disc : 1.0f);
  const float d_near = fmaxf(-bq - sq, NEARP);
  const float d_far  = fminf(-bq + sq, FARP);
  const bool  mask_bound = hit && (d_near < d_far);

  float lev[NL];
#pragma unroll
  for (int l = 0; l < NL; ++l) lev[l] = levels[l];
  const float b2v = b2[0];

  // B-matrix (4x16 f32): VGPR0 = row K0 (lanes 0-15) / K2 (16-31),
  //                      VGPR1 = row K1 (lanes 0-15) / K3==bias (16-31).
  v2f   Bw[4];
  float w2r[4];
#pragma unroll
  for (int j = 0; j < 4; ++j) {
    const int n  = j * 16 + m;
    const float k0 = hf ? W1[2 * NH + n] : W1[0 * NH + n];
    const float k1 = hf ? b1[n]          : W1[1 * NH + n];
    Bw[j].x = k0;
    Bw[j].y = k1;
    w2r[j]  = W2[n];
  }

  const float stepInv = 1.0f / (float)(NS - 1);

  // ---------------- Phase 1: march 64 samples, WMMA MLP per 16-ray tile ---
  for (int t = 0; t < NS; ++t) {
    const float tt   = (float)t * stepInv;
    const float dsmp = d_near + (d_far - d_near) * tt;
    const float px = ox + dx * dsmp, py = oy + dy * dsmp, pz = oz + dz * dsmp;
    v2f a;
    a.x = hf ? pz : px;
    a.y = hf ? 1.0f : py;
    float sum[8];
    mlp_wmma(a, Bw, w2r, sum);
    if (lane == 0) {
#pragma unroll
      for (int v = 0; v < 8; ++v) scalLDS[w][v][t] = sum[v];
    }
    if (lane == 16) {
#pragma unroll
      for (int v = 0; v < 8; ++v) scalLDS[w][v + 8][t] = sum[v];
    }
    asm volatile("s_wait_dscnt 0x0" ::: "memory");
    if (lane < 16) {
      const float nrm  = sqrtf(px * px + py * py + pz * pz);
      const float sfin = scalLDS[w][m][t] + b2v + (R0C - nrm);
      scalLDS[w][m][t] = sfin;
      valLDS[w][m][t]  = (nrm < R_SPHERE) ? 1 : 0;
    }
  }
  asm volatile("s_wait_dscnt 0x0" ::: "memory");

  // ---------------- Phase 2: interval selection, 2 (ray,level) tasks/lane -
  float dfs[2], dbs[2], sfs[2], sbs[2], dcur[2], dini[2];
  bool  msec[2], mint[2];
#pragma unroll
  for (int task = 0; task < 2; ++task) {
    const int   l  = 2 * task + hf;
    const float Lv = lev[l];
    float best_rank = -1e30f;
    int   best_idx  = 0;
    float minv      = 1e30f;
    int   low_idx   = 0;
    bool  anyhit    = false;
    float scp = scalLDS[w][m][0];
    bool  vp  = valLDS[w][m][0] != 0;
    for (int i = 0; i < NS - 1; ++i) {
      const float scb = scalLDS[w][m][i + 1];
      const bool  vb  = valLDS[w][m][i + 1] != 0;
      const bool  in_int = (scp >= Lv) && (Lv >= scb) && vp && vb;
      anyhit = anyhit || in_int;
      const float dsb  = Lv - scb;
      const float sgn  = (dsb > 0.0f) ? 1.0f : ((dsb < 0.0f) ? -1.0f : 0.0f);
      const float rank = sgn * (float)(NS - 1 - i);
      if (rank > best_rank) { best_rank = rank; best_idx = i; }
      if (scb < minv) { minv = scb; low_idx = i; }
      scp = scb;
      vp  = vb;
    }
    const int   idx = anyhit ? best_idx : low_idx;
    const float dF  = d_near + (d_far - d_near) * ((float)idx * stepInv);
    const float dB  = d_near + (d_far - d_near) * ((float)(idx + 1) * stepInv);
    const float sF  = scalLDS[w][m][idx];
    const float sB  = scalLDS[w][m][idx + 1];
    const bool  mi  = (sF >= Lv) && (Lv >= sB);
    const float sd  = sF - sB;
    const bool  vd  = fabsf(sd) > EPSI;
    const bool  ms  = mi && vd;
    const float dsec = ((Lv - sB) * dF + (sF - Lv) * dB) / (vd ? sd : 1.0f);
    const float di   = ms ? dsec : dB;
    dfs[task] = dF; dbs[task] = dB; sfs[task] = sF; sbs[task] = sB;
    dcur[task] = di; dini[task] = di; msec[task] = ms; mint[task] = mi;
  }

  // ---------------- Phase 3: secant refinement (WMMA per level-tile) ------
#pragma unroll 1
  for (int step = 0; step < 4; ++step) {
    float snrm[2] = {0.0f, 0.0f};
#pragma unroll
    for (int gl = 0; gl < NL; ++gl) {
      const int  tsk = gl >> 1;
      const bool own = (hf == (gl & 1));
      const float dmine = dcur[tsk];
      const float doth  = __shfl_xor(dmine, 16, 32);
      const float dg    = own ? dmine : doth;
      const float px = ox + dx * dg, py = oy + dy * dg, pz = oz + dz * dg;
      const float nrm = sqrtf(px * px + py * py + pz * pz);
      if (own) snrm[tsk] = nrm;
      v2f a;
      a.x = hf ? pz : px;
      a.y = hf ? 1.0f : py;
      float sum[8];
      mlp_wmma(a, Bw, w2r, sum);
      if (lane == 0) {
#pragma unroll
        for (int v = 0; v < 8; ++v) tmpLDS[w][gl][v] = sum[v];
      }
      if (lane == 16) {
#pragma unroll
        for (int v = 0; v < 8; ++v) tmpLDS[w][gl][v + 8] = sum[v];
      }
    }
    asm volatile("s_wait_dscnt 0x0" ::: "memory");
#pragma unroll
    for (int task = 0; task < 2; ++task) {
      const int   l    = 2 * task + hf;
      const float Lv   = lev[l];
      const float smid = tmpLDS[w][l][m] + b2v + (R0C - snrm[task]);
      const bool  mv   = snrm[task] < R_SPHERE;
      const bool  upf  = (smid > Lv) && mv;
      const bool  upb  = (smid < Lv) && mv;
      if (upf) { dfs[task] = dcur[task]; sfs[task] = smid; }
      if (upb) { dbs[task] = dcur[task]; sbs[task] = smid; }
      const float sd   = sfs[task] - sbs[task];
      const bool  ok   = fabsf(sd) > EPSI;
      const float dnew = ((Lv - sbs[task]) * dfs[task] + (sfs[task] - Lv) * dbs[task]) /
                         (ok ? sd : 1.0f);
      dcur[task] = ok ? dnew : dcur[task];
    }
  }

  // ---------------- Phase 4: per-ray sort of L=4 levels, write out --------
#pragma unroll
  for (int task = 0; task < 2; ++task) {
    const int   l    = 2 * task + hf;
    const float dint = msec[task] ? dcur[task] : dini[task];
    diLDS[w][m][l] = dint;
    miLDS[w][m][l] = mint[task] ? 1.0f : 0.0f;
  }
  asm volatile("s_wait_dscnt 0x0" ::: "memory");
  if (lane < 16) {
    float dv[NL], mk[NL];
#pragma unroll
    for (int l = 0; l < NL; ++l) {
      dv[l] = diLDS[w][m][l];
      mk[l] = miLDS[w][m][l];
    }
#define CSWAP(i, j)                                        \
  if (dv[i] > dv[j]) {                                     \
    float td = dv[i]; dv[i] = dv[j]; dv[j] = td;           \
    float tm = mk[i]; mk[i] = mk[j]; mk[j] = tm;           \
  }
    CSWAP(0, 1) CSWAP(2, 3) CSWAP(0, 2) CSWAP(1, 3) CSWAP(1, 2)
#undef CSWAP
    const size_t base = (size_t)g * NL;
#pragma unroll
    for (int l = 0; l < NL; ++l) {
      out_dist[base + l] = mask_bound ? dv[l] : 0.0f;
      out_mask[base + l] = (mask_bound && mk[l] != 0.0f) ? 1.0f : 0.0f;
    }
  }
}

extern "C" void kernel_launch(void* const* d_in, const int* in_sizes, int n_in,
                              void* d_out, int out_size, void* d_ws, size_t ws_size,
                              hipStream_t stream) {
  const float* rays_o = (const float*)d_in[0];
  const float* rays_d = (const float*)d_in[1];
  const float* levels = (const float*)d_in[2];
  const float* W1     = (const float*)d_in[3];
  const float* b1     = (const float*)d_in[4];
  const float* W2     = (const float*)d_in[5];
  const float* b2     = (const float*)d_in[6];
  const int N = in_sizes[0] / 3;  // B*R rays

  float* out      = (float*)d_out;
  float* out_dist = out;                     // (B,R,L) dist
  float* out_mask = out + (size_t)N * NL;    // (B,R,L) mask

  const int blocks = (N + 127) / 128;  // 128 rays / block (8 waves x 16 rays)
  manifold_march_kernel<<<blocks, 256, 0, stream>>>(rays_o, rays_d, levels, W1, b1,
                                                    W2, b2, out_dist, out_mask, N);
}